// GeometricRegistrationLoss_43997644980346
// MI455X (gfx1250) — compile-verified
//
#include <hip/hip_runtime.h>
#include <math.h>

// ---------------- problem constants (from reference setup_inputs) ----------------
#define BATCH 16
#define IMG_H 384
#define IMG_W 1280
#define NPIX (IMG_H * IMG_W)          // 491520
#define NTOTAL (BATCH * NPIX)         // 7864320
#define EPS_F 1e-8f
#define REG_LAMBDA_F 0.1f

#define BLOCKS_PER_IMG 256
#define THREADS 256
#define NBLOCKS (BLOCKS_PER_IMG * BATCH)   // 4096
#define NPART (2 * NBLOCKS)                // 8192 partial sums (multiple of 64)

// ws layout (in floats)
#define WS_TN2C 0      // 16 * 12 : R(9) row-major + t(3), maps next-frame pts -> current frame
#define WS_TC2N 192    // 16 * 12 : maps current-frame pts -> next frame
#define WS_KINV 384    // 9
#define WS_PART 400    // NPART floats

typedef __attribute__((ext_vector_type(2))) float v2f;
typedef __attribute__((ext_vector_type(8))) float v8f;

// ---------------- kernel 1: per-batch transforms + K inverse ----------------

__device__ __forceinline__ void rot_from_aa(float wx, float wy, float wz, float R[9]) {
    float theta2 = wx * wx + wy * wy + wz * wz;
    float theta  = sqrtf(theta2 + 1e-12f);
    float A  = sinf(theta) / theta;
    float Bc = (1.0f - cosf(theta)) / (theta2 + 1e-12f);
    R[0] = 1.0f + Bc * (-(wy * wy + wz * wz));
    R[1] = -A * wz + Bc * (wx * wy);
    R[2] =  A * wy + Bc * (wx * wz);
    R[3] =  A * wz + Bc * (wx * wy);
    R[4] = 1.0f + Bc * (-(wx * wx + wz * wz));
    R[5] = -A * wx + Bc * (wy * wz);
    R[6] = -A * wy + Bc * (wx * wz);
    R[7] =  A * wx + Bc * (wy * wz);
    R[8] = 1.0f + Bc * (-(wx * wx + wy * wy));
}

// T_rel = inv(Tb) @ Ta for rigid transforms: R = Rb^T Ra, t = Rb^T (pa - pb)
__device__ __forceinline__ void store_rel(float* o, const float Ra[9], const float pa[3],
                                          const float Rb[9], const float pb[3]) {
    for (int i = 0; i < 3; ++i)
        for (int j = 0; j < 3; ++j)
            o[3 * i + j] = Rb[i] * Ra[j] + Rb[3 + i] * Ra[3 + j] + Rb[6 + i] * Ra[6 + j];
    float dx = pa[0] - pb[0], dy = pa[1] - pb[1], dz = pa[2] - pb[2];
    o[9]  = Rb[0] * dx + Rb[3] * dy + Rb[6] * dz;
    o[10] = Rb[1] * dx + Rb[4] * dy + Rb[7] * dz;
    o[11] = Rb[2] * dx + Rb[5] * dy + Rb[8] * dz;
}

__global__ void grl_setup_kernel(const float* __restrict__ cur_pos,
                                 const float* __restrict__ nxt_pos,
                                 const float* __restrict__ cur_ang,
                                 const float* __restrict__ nxt_ang,
                                 const float* __restrict__ Kmat,
                                 float* __restrict__ ws) {
    int b = threadIdx.x;
    if (b < BATCH) {
        float Rc[9], Rn[9];
        rot_from_aa(cur_ang[3 * b], cur_ang[3 * b + 1], cur_ang[3 * b + 2], Rc);
        rot_from_aa(nxt_ang[3 * b], nxt_ang[3 * b + 1], nxt_ang[3 * b + 2], Rn);
        float pc[3] = {cur_pos[3 * b], cur_pos[3 * b + 1], cur_pos[3 * b + 2]};
        float pn[3] = {nxt_pos[3 * b], nxt_pos[3 * b + 1], nxt_pos[3 * b + 2]};
        // T_n2c = inv(T_current) @ T_next   (a = next, b = current)
        store_rel(ws + WS_TN2C + 12 * b, Rn, pn, Rc, pc);
        // T_c2n = inv(T_next) @ T_current   (a = current, b = next)
        store_rel(ws + WS_TC2N + 12 * b, Rc, pc, Rn, pn);
    }
    if (threadIdx.x == 0) {
        float m00 = Kmat[0], m01 = Kmat[1], m02 = Kmat[2];
        float m10 = Kmat[3], m11 = Kmat[4], m12 = Kmat[5];
        float m20 = Kmat[6], m21 = Kmat[7], m22 = Kmat[8];
        float det = m00 * (m11 * m22 - m12 * m21) - m01 * (m10 * m22 - m12 * m20)
                  + m02 * (m10 * m21 - m11 * m20);
        float id = 1.0f / det;
        float* ki = ws + WS_KINV;
        ki[0] = (m11 * m22 - m12 * m21) * id;
        ki[1] = (m02 * m21 - m01 * m22) * id;
        ki[2] = (m01 * m12 - m02 * m11) * id;
        ki[3] = (m12 * m20 - m10 * m22) * id;
        ki[4] = (m00 * m22 - m02 * m20) * id;
        ki[5] = (m02 * m10 - m00 * m12) * id;
        ki[6] = (m10 * m21 - m11 * m20) * id;
        ki[7] = (m01 * m20 - m00 * m21) * id;
        ki[8] = (m00 * m11 - m01 * m10) * id;
    }
}

// ---------------- kernel 2: dual-direction warp + per-block partial sums ----------------

__device__ __forceinline__ float bilin_tap(const float* __restrict__ src,
                                           float xf, float yf, float wgt) {
    bool valid = (xf >= 0.0f) && (xf <= (float)(IMG_W - 1)) &&
                 (yf >= 0.0f) && (yf <= (float)(IMG_H - 1));
    float xc = fminf(fmaxf(xf, 0.0f), (float)(IMG_W - 1));
    float yc = fminf(fmaxf(yf, 0.0f), (float)(IMG_H - 1));
    int xi = (int)xc;
    int yi = (int)yc;
    float val = src[yi * IMG_W + xi];           // stays L2-resident (63 MB << 192 MB L2)
    return valid ? val * wgt : 0.0f;
}

__device__ __forceinline__ float warp_residual(
    const float* __restrict__ src, float d,
    float rx, float ry, float rz,
    float r00, float r01, float r02, float r10, float r11, float r12,
    float r20, float r21, float r22, float t0, float t1, float t2,
    float k00, float k01, float k02, float k10, float k11, float k12,
    float k20, float k21, float k22) {
    // back-project dst pixel, transform into src frame, re-project
    float px = d * rx, py = d * ry, pz = d * rz;
    float sx = r00 * px + r01 * py + r02 * pz + t0;
    float sy = r10 * px + r11 * py + r12 * pz + t1;
    float sz = r20 * px + r21 * py + r22 * pz + t2;
    float q0 = k00 * sx + k01 * sy + k02 * sz;
    float q1 = k10 * sx + k11 * sy + k12 * sz;
    float q2 = k20 * sx + k21 * sy + k22 * sz;
    float scale = (fabsf(q2) > EPS_F) ? (1.0f / q2) : 1.0f;
    float x = q0 * scale;
    float y = q1 * scale;
    // replicate reference's normalize -> denormalize round trip
    float xn = 2.0f * x / (float)(IMG_W - 1) - 1.0f;
    float yn = 2.0f * y / (float)(IMG_H - 1) - 1.0f;
    float ix = (xn + 1.0f) * (float)(IMG_W - 1) * 0.5f;
    float iy = (yn + 1.0f) * (float)(IMG_H - 1) * 0.5f;
    float x0 = floorf(ix), y0 = floorf(iy);
    float wx1 = ix - x0, wy1 = iy - y0;
    float v = bilin_tap(src, x0,        y0,        (1.0f - wx1) * (1.0f - wy1))
            + bilin_tap(src, x0 + 1.0f, y0,        wx1 * (1.0f - wy1))
            + bilin_tap(src, x0,        y0 + 1.0f, (1.0f - wx1) * wy1)
            + bilin_tap(src, x0 + 1.0f, y0 + 1.0f, wx1 * wy1);
    return fabsf(v - d);
}

__global__ __launch_bounds__(THREADS) void grl_warp_loss_kernel(
    const float* __restrict__ cur, const float* __restrict__ nxt,
    const float* __restrict__ Kmat, const float* __restrict__ ws,
    float* __restrict__ partials) {
    const int b = blockIdx.y;                       // batch is uniform per block
    const float* __restrict__ img_cur = cur + (size_t)b * NPIX;
    const float* __restrict__ img_nxt = nxt + (size_t)b * NPIX;

    // uniform per-block constants -> scalar regs
    const float k00 = Kmat[0], k01 = Kmat[1], k02 = Kmat[2];
    const float k10 = Kmat[3], k11 = Kmat[4], k12 = Kmat[5];
    const float k20 = Kmat[6], k21 = Kmat[7], k22 = Kmat[8];
    const float* ki = ws + WS_KINV;
    const float i00 = ki[0], i01 = ki[1], i02 = ki[2];
    const float i10 = ki[3], i11 = ki[4], i12 = ki[5];
    const float i20 = ki[6], i21 = ki[7], i22 = ki[8];
    const float* Ta = ws + WS_TN2C + 12 * b;  // next pts -> current frame (src = cur)
    const float a00 = Ta[0], a01 = Ta[1], a02 = Ta[2], a10 = Ta[3], a11 = Ta[4], a12 = Ta[5];
    const float a20 = Ta[6], a21 = Ta[7], a22 = Ta[8], at0 = Ta[9], at1 = Ta[10], at2 = Ta[11];
    const float* Tc = ws + WS_TC2N + 12 * b;  // current pts -> next frame (src = nxt)
    const float c00 = Tc[0], c01 = Tc[1], c02 = Tc[2], c10 = Tc[3], c11 = Tc[4], c12 = Tc[5];
    const float c20 = Tc[6], c21 = Tc[7], c22 = Tc[8], ct0 = Tc[9], ct1 = Tc[10], ct2 = Tc[11];

    float s_next = 0.0f;   // |warp(cur -> next grid) - next|
    float s_prev = 0.0f;   // |warp(nxt -> current grid) - cur|

    for (int p = blockIdx.x * blockDim.x + threadIdx.x; p < NPIX;
         p += BLOCKS_PER_IMG * THREADS) {
        int h = p / IMG_W;
        int w = p - h * IMG_W;
        float fu = (float)w, fv = (float)h;
        float rx = i00 * fu + i01 * fv + i02;
        float ry = i10 * fu + i11 * fv + i12;
        float rz = i20 * fu + i21 * fv + i22;
        float dn = img_nxt[p];   // coalesced
        float dc = img_cur[p];   // coalesced
        s_next += warp_residual(img_cur, dn, rx, ry, rz,
                                a00, a01, a02, a10, a11, a12, a20, a21, a22, at0, at1, at2,
                                k00, k01, k02, k10, k11, k12, k20, k21, k22);
        s_prev += warp_residual(img_nxt, dc, rx, ry, rz,
                                c00, c01, c02, c10, c11, c12, c20, c21, c22, ct0, ct1, ct2,
                                k00, k01, k02, k10, k11, k12, k20, k21, k22);
    }

    // wave32 reduce, then cross-wave via LDS
    for (int off = 16; off > 0; off >>= 1) {
        s_next += __shfl_down(s_next, off, 32);
        s_prev += __shfl_down(s_prev, off, 32);
    }
    __shared__ float red[16];
    int lane = threadIdx.x & 31;
    int wid = threadIdx.x >> 5;
    if (lane == 0) { red[wid] = s_next; red[8 + wid] = s_prev; }
    __syncthreads();
    if (threadIdx.x == 0) {
        float sn = 0.0f, sp = 0.0f;
        for (int i = 0; i < 8; ++i) { sn += red[i]; sp += red[8 + i]; }
        int gid = blockIdx.y * gridDim.x + blockIdx.x;
        partials[2 * gid]     = sn;   // plain stores -> bitwise-deterministic reduction
        partials[2 * gid + 1] = sp;
    }
}

// ---------------- kernel 3: WMMA-based deterministic final reduction ----------------
// One wave32. D = A x Ones + C accumulates per-row sums of 16x4 f32 chunks; the
// N=0 column of the 16x16 f32 C/D layout lives in lanes 0 (M=0..7) and 16 (M=8..15),
// so summing those lanes' 8 accumulators gives the grand total.

__global__ void grl_final_reduce_kernel(const float* __restrict__ partials,
                                        float* __restrict__ out) {
    int lane = threadIdx.x;   // 0..31, EXEC all ones (WMMA requirement)
    v8f c = {};
    v2f bones;
    bones.x = 1.0f;
    bones.y = 1.0f;
    for (int chunk = 0; chunk < NPART / 64; ++chunk) {
        v2f a;
        a.x = partials[chunk * 64 + lane];        // any lane<->element bijection works:
        a.y = partials[chunk * 64 + 32 + lane];   // B == ones => D row-sums => total sum
        c = __builtin_amdgcn_wmma_f32_16x16x4_f32(
            /*neg_a=*/false, a, /*neg_b=*/false, bones,
            /*c_mod=*/(short)0, c, /*reuse_a=*/false, /*reuse_b=*/false);
    }
    float lane_tot = c[0] + c[1] + c[2] + c[3] + c[4] + c[5] + c[6] + c[7];
    float t0  = __shfl(lane_tot, 0, 32);
    float t16 = __shfl(lane_tot, 16, 32);
    if (lane == 0) {
        // (mean_prev + mean_next)/2 * lambda == (sum_prev + sum_next) * lambda / (2N)
        out[0] = (t0 + t16) * (REG_LAMBDA_F / (2.0f * (float)NTOTAL));
    }
}

// ---------------- launcher ----------------

extern "C" void kernel_launch(void* const* d_in, const int* in_sizes, int n_in,
                              void* d_out, int out_size, void* d_ws, size_t ws_size,
                              hipStream_t stream) {
    const float* cur_depth = (const float*)d_in[0];
    const float* nxt_depth = (const float*)d_in[1];
    const float* cur_pos   = (const float*)d_in[2];
    const float* nxt_pos   = (const float*)d_in[3];
    const float* cur_ang   = (const float*)d_in[4];
    const float* nxt_ang   = (const float*)d_in[5];
    const float* Kmat      = (const float*)d_in[6];
    float* ws  = (float*)d_ws;
    float* out = (float*)d_out;

    grl_setup_kernel<<<1, 32, 0, stream>>>(cur_pos, nxt_pos, cur_ang, nxt_ang, Kmat, ws);

    dim3 grid(BLOCKS_PER_IMG, BATCH);
    grl_warp_loss_kernel<<<grid, THREADS, 0, stream>>>(cur_depth, nxt_depth, Kmat, ws,
                                                       ws + WS_PART);

    grl_final_reduce_kernel<<<1, 32, 0, stream>>>(ws + WS_PART, out);
}